// LightGCN_26156350832667
// MI455X (gfx1250) — compile-verified
//
#include <hip/hip_runtime.h>
#include <stdint.h>

#define DIM 64
#define WAVE 32
#define BLOCK_THREADS 256
#define WAVES_PER_BLOCK (BLOCK_THREADS / WAVE)
#define EDGES_PER_WAVE 256
#define BLOCK_EDGES (WAVES_PER_BLOCK * EDGES_PER_WAVE) /* 2048 edges/chunk */
#define MAX_SPMM_BLOCKS 1024

typedef __attribute__((ext_vector_type(4))) unsigned int v4u;
typedef __attribute__((ext_vector_type(4))) int          v4i;
typedef __attribute__((ext_vector_type(8))) int          v8i;

#if __has_builtin(__builtin_amdgcn_tensor_load_to_lds) && \
    __has_builtin(__builtin_amdgcn_s_wait_tensorcnt)
#define USE_TDM 1
#else
#define USE_TDM 0
#endif

__device__ __forceinline__ void atom_add_f32(float* p, float v) {
  // lowers to global_atomic_add_f32 scope:SCOPE_DEV (verified round 1)
  unsafeAtomicAdd(p, v);
}

#if USE_TDM
// 1D tensor-DMA: copy n_valid 4B elements (zero-fill up to n_tile) global -> LDS.
// D# per CDNA5 ISA: group0 = {count=1, lds_addr,
// global_addr[56:0], type=2}; group1 = {data_size=4B, tensor_dim0=n_valid,
// tensor_dim1=1, tile_dim0=n_tile, tile_dim1=1}. OOB reads return zero -> tail-safe.
__device__ __forceinline__ void tdm_load_1d(uint32_t lds_off, const void* g,
                                            uint32_t n_valid, uint32_t n_tile) {
  uint64_t ga = (uint64_t)(uintptr_t)g;
  v4u g0;
  g0[0] = 1u;                                                // count=1 (user descriptor)
  g0[1] = lds_off;                                           // lds_addr (bytes)
  g0[2] = (uint32_t)ga;                                      // global_addr[31:0]
  g0[3] = ((uint32_t)(ga >> 32) & 0x01FFFFFFu) | (2u << 30); // addr[56:32] | type=2
  v8i g1;
  g1[0] = (int)(2u << 16);                                   // data_size=2 (4 bytes), mask=0
  g1[1] = (int)((n_valid & 0xFFFFu) << 16);                  // tensor_dim0[15:0] @ 63:48
  g1[2] = (int)(((n_valid >> 16) & 0xFFFFu) | (1u << 16));   // tensor_dim0[31:16] | tensor_dim1=1
  g1[3] = (int)((n_tile & 0xFFFFu) << 16);                   // tile_dim0 @ 127:112
  g1[4] = 1;                                                 // tile_dim1=1, tile_dim2=0
  g1[5] = (int)n_tile;                                       // tensor_dim0_stride (don't care)
  g1[6] = 0;
  g1[7] = 0;
  v4i z4 = {0, 0, 0, 0};
#if __clang_major__ >= 23
  v8i z8 = {0, 0, 0, 0, 0, 0, 0, 0};
  __builtin_amdgcn_tensor_load_to_lds(g0, g1, z4, z4, z8, 0);
#else
  __builtin_amdgcn_tensor_load_to_lds(g0, g1, z4, z4, 0);
#endif
}
#endif

// ---------------- degree: deg[row[e]] += w[e] ----------------
__global__ void __launch_bounds__(256)
k_deg(const int* __restrict__ row, const float* __restrict__ w,
      float* __restrict__ deg, int nE) {
  int e = blockIdx.x * blockDim.x + threadIdx.x;
  if (e < nE) atom_add_f32(&deg[row[e]], w[e]);
}

// ---------------- per-edge symmetric norm ----------------
__global__ void __launch_bounds__(256)
k_norm(const int* __restrict__ row, const int* __restrict__ col,
       const float* __restrict__ w, const float* __restrict__ deg,
       float* __restrict__ nrm, int nE) {
  int e = blockIdx.x * blockDim.x + threadIdx.x;
  if (e < nE) {
    float dr = deg[row[e]];
    float dc = deg[col[e]];
    float ir = dr > 0.0f ? rsqrtf(dr) : 0.0f;
    float ic = dc > 0.0f ? rsqrtf(dc) : 0.0f;
    nrm[e] = ir * w[e] * ic;
  }
}

// ---------------- SpMM: x_out[row] += norm * x_in[col] ----------------
// Persistent blocks, double-buffered TDM staging: wave 0 issues the next
// chunk's 3 DMAs into buf^1, waits tensorcnt<=3 (TDM completes in-order per
// wave => current buffer's DMAs done), barrier, all 8 waves compute from buf
// while the TDM engine fills buf^1. 16 lanes x float4 = one 64-float row,
// 2 edges per wave step; scatter via 4 HW f32 atomics/lane (L2-resident).
__global__ void __launch_bounds__(BLOCK_THREADS)
k_spmm(const int* __restrict__ row, const int* __restrict__ col,
       const float* __restrict__ nrm, const float* __restrict__ x_in,
       float* __restrict__ x_out, int nE, int nChunks) {
  __shared__ int   s_row[2][BLOCK_EDGES];
  __shared__ int   s_col[2][BLOCK_EDGES];
  __shared__ float s_nrm[2][BLOCK_EDGES];

  const int tid    = threadIdx.x;
  const int stride = gridDim.x;
  int chunk = blockIdx.x;
  int buf   = 0;

#if USE_TDM
  if (chunk < nChunks && tid < WAVE) {   // prologue: stage first chunk into buf 0
    const int base = chunk * BLOCK_EDGES;
    int cn = nE - base; if (cn > BLOCK_EDGES) cn = BLOCK_EDGES;
    tdm_load_1d((uint32_t)(uintptr_t)(void*)s_row[0], row + base, (uint32_t)cn, BLOCK_EDGES);
    tdm_load_1d((uint32_t)(uintptr_t)(void*)s_col[0], col + base, (uint32_t)cn, BLOCK_EDGES);
    tdm_load_1d((uint32_t)(uintptr_t)(void*)s_nrm[0], nrm + base, (uint32_t)cn, BLOCK_EDGES);
  }
#endif

  const int lane = tid & (WAVE - 1);
  const int wv   = tid >> 5;
  const int half = lane >> 4;   // which edge of the pair this lane serves
  const int q    = lane & 15;   // float4 slot within the 64-float row
  const int ebase = wv * EDGES_PER_WAVE;

  for (; chunk < nChunks; chunk += stride) {
    const int base = chunk * BLOCK_EDGES;
    int cn = nE - base; if (cn > BLOCK_EDGES) cn = BLOCK_EDGES;
    const int nxt = chunk + stride;

#if USE_TDM
    if (tid < WAVE) {
      if (nxt < nChunks) {      // stage next chunk into buf^1 while we compute buf
        const int nb = nxt * BLOCK_EDGES;
        int ncn = nE - nb; if (ncn > BLOCK_EDGES) ncn = BLOCK_EDGES;
        const int ob = buf ^ 1;
        tdm_load_1d((uint32_t)(uintptr_t)(void*)s_row[ob], row + nb, (uint32_t)ncn, BLOCK_EDGES);
        tdm_load_1d((uint32_t)(uintptr_t)(void*)s_col[ob], col + nb, (uint32_t)ncn, BLOCK_EDGES);
        tdm_load_1d((uint32_t)(uintptr_t)(void*)s_nrm[ob], nrm + nb, (uint32_t)ncn, BLOCK_EDGES);
        __builtin_amdgcn_s_wait_tensorcnt(3);  // oldest 3 (current buf) complete
      } else {
        __builtin_amdgcn_s_wait_tensorcnt(0);
      }
    }
    __syncthreads();
#else
    for (int i = tid; i < cn; i += BLOCK_THREADS) {
      s_row[buf][i] = row[base + i];
      s_col[buf][i] = col[base + i];
      s_nrm[buf][i] = nrm[base + i];
    }
    __syncthreads();
#endif

    const int*   cr = s_row[buf];
    const int*   cc = s_col[buf];
    const float* cw = s_nrm[buf];

    if (cn == BLOCK_EDGES) {
      // fast path: no per-edge bounds check (all chunks except the global tail)
      for (int t = 0; t < EDGES_PER_WAVE; t += 2) {
        const int e = ebase + t + half;
        const int   c  = cc[e];
        const int   r  = cr[e];
        const float nm = cw[e];
        const int ep = e + 16;
        if (q == 0 && ep < BLOCK_EDGES) {  // prefetch a future gather row (2 lines)
          const float* pf = x_in + (size_t)cc[ep] * DIM;
          __builtin_prefetch(pf, 0, 0);
          __builtin_prefetch(pf + 32, 0, 0);
        }
        const float4 v = *(const float4*)(x_in + (size_t)c * DIM + q * 4);
        float* o = x_out + (size_t)r * DIM + q * 4;
        atom_add_f32(o + 0, nm * v.x);
        atom_add_f32(o + 1, nm * v.y);
        atom_add_f32(o + 2, nm * v.z);
        atom_add_f32(o + 3, nm * v.w);
      }
    } else {
      for (int t = 0; t < EDGES_PER_WAVE; t += 2) {
        const int e = ebase + t + half;
        if (e < cn) {
          const int   c  = cc[e];
          const int   r  = cr[e];
          const float nm = cw[e];
          const float4 v = *(const float4*)(x_in + (size_t)c * DIM + q * 4);
          float* o = x_out + (size_t)r * DIM + q * 4;
          atom_add_f32(o + 0, nm * v.x);
          atom_add_f32(o + 1, nm * v.y);
          atom_add_f32(o + 2, nm * v.z);
          atom_add_f32(o + 3, nm * v.w);
        }
      }
    }
    __syncthreads();  // nobody may still read buf when next iteration's TDM overwrites it
    buf ^= 1;
  }
}

// ---------------- elementwise helpers (float4) ----------------
__global__ void __launch_bounds__(256)
k_copy4(float4* __restrict__ o, const float4* __restrict__ a, int n4) {
  int i = blockIdx.x * blockDim.x + threadIdx.x;
  if (i < n4) o[i] = a[i];
}
__global__ void __launch_bounds__(256)
k_add4(float4* __restrict__ o, const float4* __restrict__ a, int n4) {
  int i = blockIdx.x * blockDim.x + threadIdx.x;
  if (i < n4) {
    float4 x = o[i], y = a[i];
    x.x += y.x; x.y += y.y; x.z += y.z; x.w += y.w;
    o[i] = x;
  }
}
__global__ void __launch_bounds__(256)
k_scale4(float4* __restrict__ o, float s, int n4) {
  int i = blockIdx.x * blockDim.x + threadIdx.x;
  if (i < n4) {
    float4 x = o[i];
    x.x *= s; x.y *= s; x.z *= s; x.w *= s;
    o[i] = x;
  }
}

extern "C" void kernel_launch(void* const* d_in, const int* in_sizes, int n_in,
                              void* d_out, int out_size, void* d_ws, size_t ws_size,
                              hipStream_t stream) {
  const int*   eidx = (const int*)d_in[0];    // [2, 2E] int32
  const float* w    = (const float*)d_in[1];  // [2E]
  const float* emb  = (const float*)d_in[2];  // [N, 64]
  const int nE = in_sizes[1];                 // 2E directed edges
  const int ND = in_sizes[2];                 // N * 64
  const int N  = ND / DIM;
  const int* row = eidx;
  const int* col = eidx + nE;
  float* out = (float*)d_out;

  // workspace carve-out (256B aligned): deg | norm | xA | xB  (~94 MB)
  uintptr_t p = (uintptr_t)d_ws;
  auto carve = [&](size_t bytes) -> float* {
    float* r = (float*)p;
    p += (bytes + 255) & ~(size_t)255;
    return r;
  };
  float* deg = carve((size_t)N * 4);
  float* nrm = carve((size_t)nE * 4);
  float* xA  = carve((size_t)ND * 4);
  float* xB  = carve((size_t)ND * 4);

  hipMemsetAsync(deg, 0, (size_t)N * 4, stream);
  {
    const int th = 256, bl = (nE + th - 1) / th;
    k_deg <<<bl, th, 0, stream>>>(row, w, deg, nE);
    k_norm<<<bl, th, 0, stream>>>(row, col, w, deg, nrm, nE);
  }

  const int n4  = ND / 4;
  const int thr = 256, ble = (n4 + thr - 1) / thr;
  k_copy4<<<ble, thr, 0, stream>>>((float4*)out, (const float4*)emb, n4);  // acc = x0

  const int nChunks = (nE + BLOCK_EDGES - 1) / BLOCK_EDGES;
  const int sb = nChunks < MAX_SPMM_BLOCKS ? nChunks : MAX_SPMM_BLOCKS;
  const float* xin = emb;
  float* xcur = xA;
  float* xoth = xB;
  for (int l = 0; l < 3; ++l) {
    hipMemsetAsync(xcur, 0, (size_t)ND * 4, stream);
    k_spmm<<<sb, BLOCK_THREADS, 0, stream>>>(row, col, nrm, xin, xcur, nE, nChunks);
    k_add4<<<ble, thr, 0, stream>>>((float4*)out, (const float4*)xcur, n4);  // acc += x_l
    xin = xcur;
    float* t = xcur; xcur = xoth; xoth = t;
  }
  k_scale4<<<ble, thr, 0, stream>>>((float4*)out, 0.25f, n4);  // mean over 4 terms
}